// ensembledSeqLabel_24833500905527
// MI455X (gfx1250) — compile-verified
//
#include <hip/hip_runtime.h>
#include <stdint.h>

// Viterbi (max-plus) forward scan + in-LDS backtrack for MI455X (gfx1250).
// B=32, S=512, T=128. Memory-bound: streams 1.07 GB of transition tiles once.
// CDNA5 Tensor Data Mover (TENSOR_LOAD_TO_LDS / TENSORcnt) triple-buffers
// 64KB tiles into LDS two steps ahead of the 8 compute waves. Inner loop
// reads tiles with ds_load_b128 (4 j's per thread) to stay under the VALU
// floor on LDS instruction issue.

typedef __attribute__((ext_vector_type(4))) unsigned int v4u;
typedef __attribute__((ext_vector_type(8))) int          v8i;
typedef __attribute__((ext_vector_type(4))) int          v4i;

constexpr int kT   = 128;
constexpr int kS   = 512;
constexpr int kSP1 = 513;
constexpr unsigned kTileBytes = kT * kT * 4; // 65536

// Issue one TDM 2-D tile load: 128x128 fp32 tile, contiguous rows (stride 128
// elements), global -> LDS. Descriptor per CDNA5 ISA D# groups 0/1 (§8.3-8.4).
__device__ __forceinline__ void tdm_load_tile(const float* gsrc, unsigned lds_off)
{
    uint64_t ga = (uint64_t)(uintptr_t)gsrc;
    v4u g0;
    g0.x = 1u;                                   // count=1 (valid), user mode
    g0.y = lds_off;                              // lds_addr [63:32]
    g0.z = (unsigned)(ga & 0xFFFFFFFFu);         // global_addr [95:64]
    g0.w = ((unsigned)(ga >> 32) & 0x01FFFFFFu)  // global_addr [120:96]
         | 0x80000000u;                          // type=2 ("image") [127:126]

    v8i g1 = {
        0x00020000,                 // wg_mask=0 (not in cluster), data_size=2 (4B)
        (int)(128u << 16),          // tensor_dim0 low16 = 128 (bits 48..63)
        (int)(128u << 16),          // tensor_dim0 hi = 0; tensor_dim1 low16 = 128
        (int)(128u << 16),          // tensor_dim1 hi = 0; tile_dim0 = 128
        128,                        // tile_dim1 = 128, tile_dim2 = 0
        128,                        // tensor_dim0_stride = 128 (elements)
        (int)(16384u << 16),        // tensor_dim1_stride low16 (unused, 2D tile)
        0                           // tensor_dim1_stride hi
    };
    v4i z4 = {0, 0, 0, 0};          // groups 2/3 unused for 2-D tensors
    v8i z8 = {0, 0, 0, 0, 0, 0, 0, 0};
    __builtin_amdgcn_tensor_load_to_lds(g0, g1, z4, z4, z8, 0);
}

__global__ __launch_bounds__(256)
void viterbi_fwd_bt(const float* __restrict__ c0,   // [B, S+1, T, T]
                    const float* __restrict__ c1,   // [B, S+1, T]
                    const int*   __restrict__ sofp,
                    const int*   __restrict__ eofp,
                    int*         __restrict__ dec)  // [B, S] int32
{
    __shared__ float         tiles[3][kT][kT];     // 196608 B, TDM destinations
    __shared__ unsigned char bpbuf[kS][kT];        //  65536 B, backpointers
    __shared__ float         fsbuf[2][kT];
    __shared__ float         pv[8][kT];            // per-chunk partial maxima
    __shared__ int           pix[8][kT];           // per-chunk partial argmax

    const int b   = blockIdx.x;
    const int tid = threadIdx.x;
    const int jq  = tid & 31;                      // j-quad index: j = 4*jq..4*jq+3
    const int j0  = jq << 2;
    const int c   = tid >> 5;                      // i-chunk 0..7: i in [16c, 16c+16)
    const int i0  = c << 4;
    const int sof = sofp[0];
    const int eof = eofp[0];

    const float* c0b = c0 + (size_t)b * kSP1 * kT * kT;
    const float* c1b = c1 + (size_t)b * kSP1 * kT;

    // Step-0 init: forscores = c0[b,0,sof,:] * c1[b,0,:]   (multiply, per ref)
    if (tid < kT)
        fsbuf[0][tid] = c0b[(size_t)sof * kT + tid] * c1b[tid];

    const unsigned tiles_off = (unsigned)(uintptr_t)(void*)&tiles[0][0][0];
    const bool wave0 = (tid >> 5) == 0;            // wave-uniform predicate

    if (wave0) {                                   // prologue: 3 tiles in flight
        tdm_load_tile(c0b + (size_t)1 * kT * kT, tiles_off);
        tdm_load_tile(c0b + (size_t)2 * kT * kT, tiles_off + kTileBytes);
        tdm_load_tile(c0b + (size_t)3 * kT * kT, tiles_off + 2u * kTileBytes);
    }

    // Software-pipelined t1: value for step s held in t1cur (combiner threads).
    float t1cur = (tid < kT) ? c1b[(size_t)1 * kT + tid] : 0.0f;

    int cur = 0;
    int buf = 0;
    for (int s = 0; s < kS; ++s) {
        if (wave0)
            __builtin_amdgcn_s_wait_tensorcnt((short)2);  // tile for step s ready
        __syncthreads();   // A: tile s + fsbuf(s) visible to all waves

        // Issue next step's t1 load early; consumed next iteration.
        float t1nxt = 0.0f;
        if (tid < kT) {
            const int sn = (s + 2 <= kS) ? (s + 2) : kS;  // clamp, avoid OOB
            t1nxt = c1b[(size_t)sn * kT + tid];
        }

        const float* fs = fsbuf[cur];
        float b0 = -INFINITY, b1 = -INFINITY, b2 = -INFINITY, b3 = -INFINITY;
        int   bi0 = i0, bi1 = i0, bi2 = i0, bi3 = i0;
#pragma unroll
        for (int kk = 0; kk < 16; ++kk) {
            const int    i   = i0 + kk;
            const float  fsv = fs[i];
            const float4 tv  = *(const float4*)&tiles[buf][i][j0]; // ds_load_b128
            float v;
            v = fsv + tv.x; if (v > b0) { b0 = v; bi0 = i; }  // '>' keeps lowest i
            v = fsv + tv.y; if (v > b1) { b1 = v; bi1 = i; }
            v = fsv + tv.z; if (v > b2) { b2 = v; bi2 = i; }
            v = fsv + tv.w; if (v > b3) { b3 = v; bi3 = i; }
        }
        *(float4*)&pv[c][j0] = make_float4(b0, b1, b2, b3);   // ds_store_b128
        *(int4*)&pix[c][j0]  = make_int4(bi0, bi1, bi2, bi3); // ds_store_b128
        __syncthreads();   // B: all tile-s reads done; partials visible

        // Reuse barrier B to recycle this step's buffer for step s+3.
        if (wave0 && (s + 3) < kS)
            tdm_load_tile(c0b + (size_t)(s + 4) * kT * kT,
                          tiles_off + (unsigned)buf * kTileBytes);

        if (tid < kT) {                            // combine 8 chunks per j
            const int jj = tid;
            float best = pv[0][jj];
            int   bi   = pix[0][jj];
#pragma unroll
            for (int q = 1; q < 8; ++q) {          // ascending i: '>' keeps lowest
                const float v = pv[q][jj];
                if (v > best) { best = v; bi = pix[q][jj]; }
            }
            fsbuf[cur ^ 1][jj] = best + t1cur;
            bpbuf[s][jj]       = (unsigned char)bi;
        }
        t1cur = t1nxt;
        cur ^= 1;
        buf = (buf == 2) ? 0 : buf + 1;
    }
    __syncthreads();       // publish last step's backpointers

    // Backtrack entirely in LDS (matches ref: ptr starts at bp[S-1][eof], then
    // for t = S-1..0: ptr = bp[t][ptr]; dec[t] = ptr).
    if (tid == 0) {
        int ptr = bpbuf[kS - 1][eof];
        for (int t = kS - 1; t >= 0; --t) {
            ptr = bpbuf[t][ptr];
            dec[(size_t)b * kS + t] = ptr;
        }
    }
}

extern "C" void kernel_launch(void* const* d_in, const int* in_sizes, int n_in,
                              void* d_out, int out_size, void* d_ws, size_t ws_size,
                              hipStream_t stream)
{
    const float* c0   = (const float*)d_in[0];
    const float* c1   = (const float*)d_in[1];
    const int*   sofp = (const int*)d_in[2];
    const int*   eofp = (const int*)d_in[3];
    const int B = in_sizes[1] / (kSP1 * kT);       // 32
    viterbi_fwd_bt<<<dim3(B), dim3(256), 0, stream>>>(c0, c1, sofp, eofp, (int*)d_out);
}